// StateNetwork_67602785239168
// MI455X (gfx1250) — compile-verified
//
#include <hip/hip_runtime.h>
#include <hip/hip_bf16.h>

// Problem constants (from reference)
#define BGRAPHS 8
#define NNODES  4096
#define FEMB    128
#define HDIM    3
#define LRELU_ALPHA 0.2f
#define FC_K    (NNODES * HDIM)   // 12288
#define FC_OUT  100

typedef float v2f __attribute__((ext_vector_type(2)));
typedef float v8f __attribute__((ext_vector_type(8)));

#if defined(__has_builtin)
#if __has_builtin(__builtin_amdgcn_wmma_f32_16x16x4_f32)
#define HAVE_WMMA_F32_16X16X4 1
#endif
#endif

// ---------------------------------------------------------------------------
// Kernel 1: C = emb[4096x128] @ Baug[128x16] via V_WMMA_F32_16X16X4_F32.
// Baug cols: 0..2 = W, 3 = W@a[:3] (-> e_src), 4 = W@a[3:] (-> e_dst), rest 0.
// One wave computes one 16-row tile (K=128 -> 32 WMMA steps). 8 waves/block,
// 32 blocks cover all 4096 rows. Results repacked through LDS into
// wh4[m] = (Wh0, Wh1, Wh2, e_dst[m]) and esrc[m].
// ---------------------------------------------------------------------------
#ifdef HAVE_WMMA_F32_16X16X4
__global__ __launch_bounds__(256) void k_wh_wmma(
    const float* __restrict__ emb, const float* __restrict__ W,
    const float* __restrict__ a, float4* __restrict__ wh4,
    float* __restrict__ esrc) {
  __shared__ float baug[16 * 132];       // B^T, padded pitch 132 (bank-conflict free)
  __shared__ float ctile[8][16][16];
  const int tid = threadIdx.x;

  const float a0 = a[0], a1 = a[1], a2 = a[2];
  const float a3 = a[3], a4 = a[4], a5 = a[5];
  for (int t = tid; t < 16 * 128; t += 256) {
    const int n = t >> 7, kk = t & 127;
    const float w0 = W[kk * 3 + 0], w1 = W[kk * 3 + 1], w2 = W[kk * 3 + 2];
    float v = 0.f;
    if (n == 0) v = w0;
    else if (n == 1) v = w1;
    else if (n == 2) v = w2;
    else if (n == 3) v = w0 * a0 + w1 * a1 + w2 * a2;   // e_src projection
    else if (n == 4) v = w0 * a3 + w1 * a4 + w2 * a5;   // e_dst projection
    baug[n * 132 + kk] = v;
  }
  __syncthreads();

  const int lane = tid & 31, wave = tid >> 5;
  const int half = lane >> 4, nn = lane & 15;
  const int m = blockIdx.x * 128 + wave * 16 + nn;      // A-matrix row (M = lane%16)
  const float* __restrict__ arow = emb + (size_t)m * FEMB;

  v8f c = {0.f, 0.f, 0.f, 0.f, 0.f, 0.f, 0.f, 0.f};
#pragma unroll 4
  for (int k0 = 0; k0 < FEMB; k0 += 4) {
    const int k = k0 + 2 * half;                        // ISA 16x4 f32 A layout
    v2f av;
    av.x = arow[k];
    av.y = arow[k + 1];
    v2f bv;
    bv.x = baug[nn * 132 + k];
    bv.y = baug[nn * 132 + k + 1];
    c = __builtin_amdgcn_wmma_f32_16x16x4_f32(false, av, false, bv,
                                              (short)0, c, false, false);
  }
#pragma unroll
  for (int r = 0; r < 8; ++r)                           // C: M = r + 8*half, N = nn
    ctile[wave][r + 8 * half][nn] = c[r];
  __syncthreads();

  if (tid < 128) {
    const int w = tid >> 4, lm = tid & 15;
    const int gm = blockIdx.x * 128 + tid;
    const float h0 = ctile[w][lm][0], h1 = ctile[w][lm][1], h2 = ctile[w][lm][2];
    const float es = ctile[w][lm][3], ed = ctile[w][lm][4];
    wh4[gm] = make_float4(h0, h1, h2, ed);
    esrc[gm] = es;
  }
}
#else
__global__ __launch_bounds__(256) void k_wh_scalar(
    const float* __restrict__ emb, const float* __restrict__ W,
    const float* __restrict__ a, float4* __restrict__ wh4,
    float* __restrict__ esrc) {
  const int m = blockIdx.x * blockDim.x + threadIdx.x;
  if (m >= NNODES) return;
  float h0 = 0.f, h1 = 0.f, h2 = 0.f;
  const float* __restrict__ row = emb + (size_t)m * FEMB;
  for (int k = 0; k < FEMB; ++k) {
    const float x = row[k];
    h0 += x * W[k * 3 + 0];
    h1 += x * W[k * 3 + 1];
    h2 += x * W[k * 3 + 2];
  }
  const float es = h0 * a[0] + h1 * a[1] + h2 * a[2];
  const float ed = h0 * a[3] + h1 * a[4] + h2 * a[5];
  wh4[m] = make_float4(h0, h1, h2, ed);
  esrc[m] = es;
}
#endif

// ---------------------------------------------------------------------------
// Kernel 2: meanWh = (1/N) * sum_j Wh_j  (uniform-softmax fallback for rows
// with no neighbors, matching the reference's all-NEG_INF softmax).
// ---------------------------------------------------------------------------
__global__ __launch_bounds__(256) void k_meanwh(const float4* __restrict__ wh4,
                                                float* __restrict__ meanwh) {
  float s0 = 0.f, s1 = 0.f, s2 = 0.f;
  for (int i = threadIdx.x; i < NNODES; i += 256) {
    const float4 v = wh4[i];
    s0 += v.x; s1 += v.y; s2 += v.z;
  }
  for (int o = 16; o > 0; o >>= 1) {
    s0 += __shfl_down(s0, o, 32);
    s1 += __shfl_down(s1, o, 32);
    s2 += __shfl_down(s2, o, 32);
  }
  __shared__ float red[8][3];
  const int lane = threadIdx.x & 31, wave = threadIdx.x >> 5;
  if (lane == 0) { red[wave][0] = s0; red[wave][1] = s1; red[wave][2] = s2; }
  __syncthreads();
  if (threadIdx.x == 0) {
    float t0 = 0.f, t1 = 0.f, t2 = 0.f;
    for (int w = 0; w < 8; ++w) { t0 += red[w][0]; t1 += red[w][1]; t2 += red[w][2]; }
    const float inv = 1.f / (float)NNODES;
    meanwh[0] = t0 * inv; meanwh[1] = t1 * inv; meanwh[2] = t2 * inv;
  }
}

// ---------------------------------------------------------------------------
// Kernel 3: fused masked softmax + attn@Wh + ELU. One block per (row i, graph
// b). Streams adj row as coalesced int4 (the 537 MB HBM bottleneck), pulls
// (Wh, e_dst) as cached float4, accumulates s = sum w and hvec = sum w*Wh,
// then h_i = elu(hvec/s). Softmax ratio is shift-invariant so no max pass.
// ---------------------------------------------------------------------------
__device__ __forceinline__ void attn_acc(int adjv, float es, const float4 w,
                                         float& s, float& h0, float& h1, float& h2) {
  float e = es + w.w;
  e = e > 0.f ? e : LRELU_ALPHA * e;
  const float ww = adjv ? __expf(e) : 0.f;
  s += ww;
  h0 += ww * w.x; h1 += ww * w.y; h2 += ww * w.z;
}

__global__ __launch_bounds__(256) void k_attn(
    const int* __restrict__ adj, const float4* __restrict__ wh4,
    const float* __restrict__ esrc, const float* __restrict__ meanwh,
    float* __restrict__ hout) {
  const int i = blockIdx.x, b = blockIdx.y, tid = threadIdx.x;
  const float es = esrc[i];
  const int4* __restrict__ arow =
      (const int4*)(adj + ((size_t)b * NNODES + i) * NNODES);

  float s = 0.f, h0 = 0.f, h1 = 0.f, h2 = 0.f;
#pragma unroll
  for (int c = 0; c < 4; ++c) {
    const int jv = c * 256 + tid;          // int4 index: j = 4*jv
    const int4 av = arow[jv];
    const float4 w0 = wh4[jv * 4 + 0];
    const float4 w1 = wh4[jv * 4 + 1];
    const float4 w2 = wh4[jv * 4 + 2];
    const float4 w3 = wh4[jv * 4 + 3];
    attn_acc(av.x, es, w0, s, h0, h1, h2);
    attn_acc(av.y, es, w1, s, h0, h1, h2);
    attn_acc(av.z, es, w2, s, h0, h1, h2);
    attn_acc(av.w, es, w3, s, h0, h1, h2);
  }

  for (int o = 16; o > 0; o >>= 1) {
    s  += __shfl_down(s,  o, 32);
    h0 += __shfl_down(h0, o, 32);
    h1 += __shfl_down(h1, o, 32);
    h2 += __shfl_down(h2, o, 32);
  }
  __shared__ float red[8][4];
  const int lane = tid & 31, wave = tid >> 5;
  if (lane == 0) { red[wave][0] = s; red[wave][1] = h0; red[wave][2] = h1; red[wave][3] = h2; }
  __syncthreads();
  if (tid == 0) {
    float S = 0.f, H0 = 0.f, H1 = 0.f, H2 = 0.f;
    for (int w = 0; w < 8; ++w) {
      S += red[w][0]; H0 += red[w][1]; H1 += red[w][2]; H2 += red[w][3];
    }
    float o0, o1, o2;
    if (S > 0.f) {
      const float inv = 1.f / S;
      o0 = H0 * inv; o1 = H1 * inv; o2 = H2 * inv;
    } else {                                    // empty row -> uniform softmax
      o0 = meanwh[0]; o1 = meanwh[1]; o2 = meanwh[2];
    }
    o0 = o0 > 0.f ? o0 : __expf(o0) - 1.f;      // ELU
    o1 = o1 > 0.f ? o1 : __expf(o1) - 1.f;
    o2 = o2 > 0.f ? o2 : __expf(o2) - 1.f;
    float* __restrict__ dst = hout + ((size_t)b * NNODES + i) * HDIM;
    dst[0] = o0; dst[1] = o1; dst[2] = o2;
  }
}

// ---------------------------------------------------------------------------
// Kernel 4: out[b][o] = h[b] . fc1_w[o] + fc1_b[o]. One block per output
// column o; fc1_w row streamed coalesced once, 8 batch accumulators/thread.
// ---------------------------------------------------------------------------
__global__ __launch_bounds__(256) void k_fc1(
    const float* __restrict__ h, const float* __restrict__ fw,
    const float* __restrict__ fb, float* __restrict__ out) {
  const int o = blockIdx.x, tid = threadIdx.x;
  const float* __restrict__ wrow = fw + (size_t)o * FC_K;
  float acc[BGRAPHS];
#pragma unroll
  for (int b = 0; b < BGRAPHS; ++b) acc[b] = 0.f;

  for (int k = tid; k < FC_K; k += 256) {
    const float wv = wrow[k];
#pragma unroll
    for (int b = 0; b < BGRAPHS; ++b) acc[b] += wv * h[b * FC_K + k];
  }
  for (int off = 16; off > 0; off >>= 1) {
#pragma unroll
    for (int b = 0; b < BGRAPHS; ++b) acc[b] += __shfl_down(acc[b], off, 32);
  }
  __shared__ float red[8][BGRAPHS];
  const int lane = tid & 31, wave = tid >> 5;
  if (lane == 0) {
#pragma unroll
    for (int b = 0; b < BGRAPHS; ++b) red[wave][b] = acc[b];
  }
  __syncthreads();
  if (tid < BGRAPHS) {
    float ssum = 0.f;
    for (int w = 0; w < 8; ++w) ssum += red[w][tid];
    out[tid * FC_OUT + o] = ssum + fb[o];
  }
}

// ---------------------------------------------------------------------------
extern "C" void kernel_launch(void* const* d_in, const int* in_sizes, int n_in,
                              void* d_out, int out_size, void* d_ws, size_t ws_size,
                              hipStream_t stream) {
  const int*   adj  = (const int*)d_in[0];
  const float* emb  = (const float*)d_in[1];
  const float* W    = (const float*)d_in[2];
  const float* a    = (const float*)d_in[3];
  const float* fw   = (const float*)d_in[4];
  const float* fb   = (const float*)d_in[5];
  float* out = (float*)d_out;

  // Workspace layout (floats): wh4[4096 float4] | esrc[4096] | meanwh[4] | h[8*12288]
  float* ws = (float*)d_ws;
  float4* wh4   = (float4*)ws;                 // 16384 floats
  float*  esrc  = ws + 16384;                  // 4096 floats
  float*  meanw = ws + 20480;                  // 4 floats
  float*  hbuf  = ws + 20484;                  // 98304 floats (~464 KB total ws use)

#ifdef HAVE_WMMA_F32_16X16X4
  k_wh_wmma<<<NNODES / 128, 256, 0, stream>>>(emb, W, a, wh4, esrc);
#else
  k_wh_scalar<<<NNODES / 256, 256, 0, stream>>>(emb, W, a, wh4, esrc);
#endif
  k_meanwh<<<1, 256, 0, stream>>>(wh4, meanw);
  dim3 g(NNODES, BGRAPHS);
  k_attn<<<g, 256, 0, stream>>>(adj, wh4, esrc, meanw, hbuf);
  k_fc1<<<FC_OUT, 256, 0, stream>>>(hbuf, fw, fb, out);
}